// MetaQDA_FB_16982300688662
// MI455X (gfx1250) — compile-verified
//
#include <hip/hip_runtime.h>
#include <hip/hip_bf16.h>
#include <math.h>

typedef __attribute__((ext_vector_type(2))) float v2f;
typedef __attribute__((ext_vector_type(8))) float v8f;
typedef __attribute__((ext_vector_type(4))) unsigned int u32x4;
typedef __attribute__((ext_vector_type(8))) int i32x8;
typedef __attribute__((ext_vector_type(4))) int i32x4;

#define REG_PARAM 0.1f

#if defined(__has_builtin)
#if __has_builtin(__builtin_amdgcn_tensor_load_to_lds)
#define HAVE_TDM 1
#endif
#endif

struct Params {
  float kap, kapS, coef_m, coef_x, sigscale, common_, halfcp, bias_const;
};

// -------- CDNA5 f32 WMMA 16x16x4: D = A(16x4) * B(4x16) + C ----------------
// A frag (2 VGPR): lane lo=M, hi half selects K+2; a.x=K(kb), a.y=K(kb+1)
// B frag (2 VGPR): lane lo=N, hi half selects K+2
// C/D (8 VGPR):    element (M = r + 8*hi, N = lo) lives in acc[r]
__device__ inline v8f wmma_f32_k4(v2f a, v2f b, v8f c) {
  return __builtin_amdgcn_wmma_f32_16x16x4_f32(
      /*neg_a=*/false, a, /*neg_b=*/false, b,
      /*c_mod=*/(short)0, c, /*reuse_a=*/false, /*reuse_b=*/false);
}

// reduce across the 16 lanes of each half of a wave32 (xor of bits 0..3)
__device__ inline float half16_reduce(float v) {
  v += __shfl_xor(v, 1, 32);
  v += __shfl_xor(v, 2, 32);
  v += __shfl_xor(v, 4, 32);
  v += __shfl_xor(v, 8, 32);
  return v;
}

// LDS address map matching the TDM pad feature: 2 DWORD pad after every 256
// DWORDs of the stream (pad_interval=7 -> 1024B, pad_amount=1 -> 2 DWORDs).
// element e of the streamed tile lives at DWORD index e + 2*(e>>8).
__device__ inline int lds_idx(int e) { return e + 2 * (e >> 8); }

#ifdef HAVE_TDM
// Issue a 2-D TDM tile load: tile_d1 rows x tile_d0 f32 elements, global row
// stride row_stride (elements), into LDS at byte offset lds_off (padded).
// D# group0/group1 bitfields per CDNA5 ISA.
// 6-arg builtin form (clang-23 / therock-10.0 headers).
__device__ inline void tdm_load_2d_f32(unsigned int lds_off, const float* gaddr,
                                       unsigned int tensor_d0, unsigned int tensor_d1,
                                       unsigned int tile_d0, unsigned int tile_d1,
                                       unsigned int row_stride) {
  unsigned long long ga = (unsigned long long)(uintptr_t)gaddr;
  u32x4 g0;
  g0.x = 1u;                                         // count=1, user descriptor
  g0.y = lds_off;                                    // lds_addr (bytes)
  g0.z = (unsigned int)(ga & 0xFFFFFFFFu);           // global_addr[31:0]
  g0.w = (unsigned int)((ga >> 32) & 0x01FFFFFFu)    // global_addr[56:32]
       | (2u << 30);                                 // type = 2 ("image")
  i32x8 g1;
  g1[0] = (int)((2u << 16)                           // data_size = 4 bytes
              | (1u << 20)                           // pad_enable
              | (7u << 22)                           // pad_interval: 256 DWORDs
              | (1u << 25));                         // pad_amount: 2 DWORDs
  g1[1] = (int)((tensor_d0 & 0xFFFFu) << 16);        // tensor_dim0[15:0]
  g1[2] = (int)(((tensor_d0 >> 16) & 0xFFFFu)        // tensor_dim0[31:16]
              | ((tensor_d1 & 0xFFFFu) << 16));      // tensor_dim1[15:0]
  g1[3] = (int)(((tensor_d1 >> 16) & 0xFFFFu)        // tensor_dim1[31:16]
              | ((tile_d0 & 0xFFFFu) << 16));        // tile_dim0
  g1[4] = (int)(tile_d1 & 0xFFFFu);                  // tile_dim1 (tile_dim2=0)
  g1[5] = (int)row_stride;                           // tensor_dim0_stride[31:0]
  g1[6] = 0;                                         // stride hi + dim1_stride
  g1[7] = 0;
  i32x4 z4 = {0, 0, 0, 0};                           // groups 2/3 unused (2-D)
  i32x8 z8 = {0, 0, 0, 0, 0, 0, 0, 0};
  __builtin_amdgcn_tensor_load_to_lds(g0, g1, z4, z4, z8, 0);
}
#endif

// ---------------------------------------------------------------------------
// 1) scalar parameters (device lgammaf for the Student-t bias constant)
__global__ void k_params(const float* __restrict__ kappa,
                         const float* __restrict__ nu,
                         Params* __restrict__ p, int D, int S) {
  float kap  = fabsf(kappa[0]) + 1e-6f;
  float nu_  = fmaxf(nu[0], (float)(D - 1) + 1e-6f);
  float kapS = kap + (float)S;
  p->kap      = kap;
  p->kapS     = kapS;
  p->coef_m   = kap / kapS;
  p->coef_x   = (float)S / kapS;
  p->sigscale = (kapS + 1.0f) / ((nu_ + (float)S - (float)D + 1.0f) * kapS);
  float common = nu_ + (float)S + 1.0f - (float)D;
  p->common_  = common;
  p->halfcp   = 0.5f * (common + (float)D);
  p->bias_const = lgammaf(0.5f * (common + (float)D)) - lgammaf(0.5f * common)
                - 0.5f * (float)D * logf(common);
}

// 2) stable position of each support row within its class (argsort-stable)
__global__ void k_pos(const int* __restrict__ labels, int* __restrict__ pos, int N) {
  int i = blockIdx.x * blockDim.x + threadIdx.x;
  if (i >= N) return;
  int li = labels[i], cnt = 0;
  for (int j = 0; j < i; ++j) cnt += (labels[j] == li) ? 1 : 0;
  pos[i] = cnt;
}

// 3) gather support rows into contiguous [C, S, D]
__global__ void k_gather(const float* __restrict__ Xs, const int* __restrict__ labels,
                         const int* __restrict__ pos, float* __restrict__ Xg,
                         int D, int S) {
  int i = blockIdx.x;
  size_t dst = ((size_t)labels[i] * S + pos[i]) * (size_t)D;
  size_t src = (size_t)i * D;
  for (int d = threadIdx.x; d < D; d += blockDim.x) Xg[dst + d] = Xs[src + d];
}

// 4) class means -> posterior means mu
__global__ void k_mu(const float* __restrict__ Xs, const int* __restrict__ labels,
                     const float* __restrict__ m, const Params* __restrict__ p,
                     float* __restrict__ mu, int N, int D, int S) {
  const int c = blockIdx.x;
  const float cm = p->coef_m, cx = p->coef_x;
  for (int d = threadIdx.x; d < D; d += blockDim.x) {
    float s = 0.0f;
    for (int i = 0; i < N; ++i)
      if (labels[i] == c) s += Xs[(size_t)i * D + d];
    mu[(size_t)c * D + d] = cm * m[d] + cx * (s / (float)S);
  }
}

// 5) base = L L^T + kap * m^T m   (L = diag(|d|) + strict-lower of triu_lower)
__global__ void k_base(const float* __restrict__ diag, const float* __restrict__ tl,
                       const float* __restrict__ m, const Params* __restrict__ p,
                       float* __restrict__ base, int D) {
  int idx = blockIdx.x * blockDim.x + threadIdx.x;
  if (idx >= D * D) return;
  int i = idx / D, j = idx % D;
  int kmax = i < j ? i : j;
  float s = 0.0f;
  for (int k = 0; k <= kmax; ++k) {
    float li = (k == i) ? fabsf(diag[i]) : tl[(size_t)i * D + k];
    float lj = (k == j) ? fabsf(diag[j]) : tl[(size_t)j * D + k];
    s = fmaf(li, lj, s);
  }
  base[idx] = s + p->kap * m[i] * m[j];
}

// 6) per-class Gram via f32 WMMA (K = S) fused with sigma assembly:
//    sigma_c = (base + Xg_c^T Xg_c - (kap+S) mu mu^T) * sigscale
__global__ void k_gram_sigma(const float* __restrict__ Xg, const float* __restrict__ base,
                             const float* __restrict__ mu, const Params* __restrict__ p,
                             float* __restrict__ sigma, int D, int S, int C) {
  const int lane = threadIdx.x & 31;
  const int wid  = blockIdx.x * (blockDim.x >> 5) + (threadIdx.x >> 5);
  const int nt   = D >> 4;
  const int tpc  = nt * nt;
  const int c    = wid / tpc;
  if (c >= C) return;
  const int rem = wid % tpc;
  const int tm = rem / nt, tn = rem % nt;
  const int lo = lane & 15, hi = lane >> 4;
  const int i0 = tm * 16, j0 = tn * 16;
  const float* Xc = Xg + (size_t)c * S * D;
  const float* ap = Xc + (size_t)(hi * 2) * D + i0 + lo;
  const float* bp = Xc + (size_t)(hi * 2) * D + j0 + lo;
  v8f acc = {};
  for (int k0 = 0; k0 < S; k0 += 4) {
    v2f a, b;
    a.x = ap[0]; a.y = ap[(size_t)D];     // A[m,k] = Xg[c][k][i0+m]
    b.x = bp[0]; b.y = bp[(size_t)D];     // B[k,n] = Xg[c][k][j0+n]
    acc = wmma_f32_k4(a, b, acc);
    ap += (size_t)4 * D;
    bp += (size_t)4 * D;
  }
  const float kapS = p->kapS, sc = p->sigscale;
  float* Sc = sigma + (size_t)c * D * D;
  const float* muc = mu + (size_t)c * D;
  for (int r = 0; r < 8; ++r) {
    int i = i0 + r + 8 * hi, j = j0 + lo;
    float v = base[(size_t)i * D + j] + acc[r] - kapS * muc[i] * muc[j];
    Sc[(size_t)i * D + j] = v * sc;
  }
}

// 7) in-place Cholesky (lower) per class; accumulates logdet(sigma)
__global__ void k_chol(float* __restrict__ M_all, float* __restrict__ logdet, int D) {
  const int c = blockIdx.x;
  float* M = M_all + (size_t)c * D * D;
  const int tid = threadIdx.x, nthr = blockDim.x;
  __shared__ float s_inv;
  float ldsum = 0.0f;
  for (int k = 0; k < D; ++k) {
    if (tid == 0) {
      float rkk = sqrtf(fmaxf(M[(size_t)k * D + k], 1e-30f));
      M[(size_t)k * D + k] = rkk;
      s_inv = 1.0f / rkk;
      ldsum += 2.0f * logf(rkk);
    }
    __syncthreads();
    float inv = s_inv;
    for (int i = k + 1 + tid; i < D; i += nthr) M[(size_t)i * D + k] *= inv;
    __syncthreads();
    for (int j = k + 1; j < D; ++j) {
      float rjk = M[(size_t)j * D + k];
      for (int i = j + tid; i < D; i += nthr)
        M[(size_t)i * D + j] = fmaf(-M[(size_t)i * D + k], rjk, M[(size_t)i * D + j]);
    }
    __syncthreads();
  }
  if (tid == 0) logdet[c] = ldsum;
}

// 8) W = R^{-1} (lower triangular inverse), upper part zeroed for dense GEMM
__global__ void k_trinv(const float* __restrict__ R_all, float* __restrict__ W_all, int D) {
  const int c = blockIdx.x;
  const float* R = R_all + (size_t)c * D * D;
  float* W = W_all + (size_t)c * D * D;
  for (int j = threadIdx.x; j < D; j += blockDim.x) {
    for (int i = 0; i < j; ++i) W[(size_t)i * D + j] = 0.0f;
    W[(size_t)j * D + j] = 1.0f / R[(size_t)j * D + j];
    for (int i = j + 1; i < D; ++i) {
      float s = 0.0f;
      for (int k = j; k < i; ++k)
        s = fmaf(R[(size_t)i * D + k], W[(size_t)k * D + j], s);
      W[(size_t)i * D + j] = -s / R[(size_t)i * D + i];
    }
  }
}

// 9) A_c = (1-REG) * W_c^T W_c + REG * I  via f32 WMMA (K starts at max(ti,tj))
__global__ void k_wtw_reg(const float* __restrict__ W, float* __restrict__ A,
                          int D, int C) {
  const int lane = threadIdx.x & 31;
  const int wid  = blockIdx.x * (blockDim.x >> 5) + (threadIdx.x >> 5);
  const int nt   = D >> 4;
  const int tpc  = nt * nt;
  const int c    = wid / tpc;
  if (c >= C) return;
  const int rem = wid % tpc;
  const int tm = rem / nt, tn = rem % nt;
  const int lo = lane & 15, hi = lane >> 4;
  const int i0 = tm * 16, j0 = tn * 16;
  const float* Wc = W + (size_t)c * D * D;
  int kstart = (tm > tn ? tm : tn) * 16;      // W[k,i]=0 for k<i
  const float* ap = Wc + (size_t)(kstart + hi * 2) * D + i0 + lo;
  const float* bp = Wc + (size_t)(kstart + hi * 2) * D + j0 + lo;
  v8f acc = {};
  for (int k0 = kstart; k0 < D; k0 += 4) {
    v2f a, b;
    a.x = ap[0]; a.y = ap[(size_t)D];        // A[m,k] = W[k, i0+m]  (W^T)
    b.x = bp[0]; b.y = bp[(size_t)D];        // B[k,n] = W[k, j0+n]
    acc = wmma_f32_k4(a, b, acc);
    ap += (size_t)4 * D;
    bp += (size_t)4 * D;
  }
  float* Ac = A + (size_t)c * D * D;
  for (int r = 0; r < 8; ++r) {
    int i = i0 + r + 8 * hi, j = j0 + lo;
    float v = (1.0f - REG_PARAM) * acc[r] + ((i == j) ? REG_PARAM : 0.0f);
    Ac[(size_t)i * D + j] = v;
  }
}

// 10) b_c = A_c mu_c
__global__ void k_bvec(const float* __restrict__ A, const float* __restrict__ mu,
                       float* __restrict__ bvec, int D) {
  const int c = blockIdx.x;
  const float* Ac = A + (size_t)c * D * D;
  const float* muc = mu + (size_t)c * D;
  for (int j = threadIdx.x; j < D; j += blockDim.x) {
    float s = 0.0f;
    for (int d = 0; d < D; ++d) s = fmaf(Ac[(size_t)d * D + j], muc[d], s);
    bvec[(size_t)c * D + j] = s;
  }
}

// 11) cc_c = mu_c^T A_c mu_c = b_c . mu_c
__global__ void k_cc(const float* __restrict__ bvec, const float* __restrict__ mu,
                     float* __restrict__ ccv, int D) {
  const int c = blockIdx.x;
  __shared__ float red[256];
  float s = 0.0f;
  for (int j = threadIdx.x; j < D; j += blockDim.x)
    s = fmaf(bvec[(size_t)c * D + j], mu[(size_t)c * D + j], s);
  red[threadIdx.x] = s;
  __syncthreads();
  for (int off = blockDim.x >> 1; off > 0; off >>= 1) {
    if (threadIdx.x < (unsigned)off) red[threadIdx.x] += red[threadIdx.x + off];
    __syncthreads();
  }
  if (threadIdx.x == 0) ccv[c] = red[0];
}

// 12) fused Mahalanobis + logits.
//     Block = one 16-query strip x 8 classes (one class per wave). The strip
//     (16 x D f32, TDM-padded layout) is staged into LDS ONCE per block by the
//     Tensor Data Mover, then every wave pulls its WMMA A-fragments from LDS
//     (conflict-free banks thanks to the pad) while streaming its own class's
//     sigma_inv from L2.
__global__ void k_mahal(const float* __restrict__ Xq, const float* __restrict__ A,
                        const float* __restrict__ bvec, const float* __restrict__ ccv,
                        const float* __restrict__ logdet, const Params* __restrict__ p,
                        float* __restrict__ out, int D, int C, int Q) {
  extern __shared__ float smem[];
  const int lane = threadIdx.x & 31;
  const int wave = threadIdx.x >> 5;
  const int q0 = blockIdx.x * 16;
  const int c  = blockIdx.y * (blockDim.x >> 5) + wave;
  const int lo = lane & 15, hi = lane >> 4;

#ifdef HAVE_TDM
  if (wave == 0) {
    // one TDM op per block: 16 rows x D cols, row stride D, pad 2 DW / 256 DW
    tdm_load_2d_f32((unsigned int)(uintptr_t)&smem[0],
                    Xq + (size_t)q0 * D,
                    (unsigned int)D, 16u, (unsigned int)D, 16u,
                    (unsigned int)D);
    __builtin_amdgcn_s_wait_tensorcnt(0);
  }
  __syncthreads();
#else
  for (int idx = threadIdx.x; idx < 16 * D; idx += blockDim.x) {
    int mrow = idx / D, k = idx - mrow * D;
    smem[lds_idx(idx)] = Xq[(size_t)(q0 + mrow) * D + k];
  }
  __syncthreads();
#endif
  if (c >= C) return;

  const float* Ac = A + (size_t)c * D * D;
  const float* bc = bvec + (size_t)c * D;
  float distAcc[8] = {0, 0, 0, 0, 0, 0, 0, 0};
  float xbAcc[8]   = {0, 0, 0, 0, 0, 0, 0, 0};
  for (int j0 = 0; j0 < D; j0 += 16) {
    v8f acc = {};
    const float* bp = Ac + (size_t)(hi * 2) * D + j0 + lo;
    int eA = lo * D + hi * 2;               // stream index of A[m=lo, k=hi*2]
    for (int k0 = 0; k0 < D; k0 += 4) {
      __builtin_prefetch(bp + (size_t)8 * D, 0, 0);
      v2f a = *(const v2f*)&smem[lds_idx(eA)];   // ds_load_b64, banks 4*lane
      v2f b;
      b.x = bp[0];
      b.y = bp[(size_t)D];
      acc = wmma_f32_k4(a, b, acc);
      bp += (size_t)4 * D;
      eA += 4;
    }
    for (int r = 0; r < 8; ++r) {
      int col = j0 + lo;
      float xq = smem[lds_idx((r + 8 * hi) * D + col)];
      distAcc[r] = fmaf(acc[r], xq, distAcc[r]);   // x^T A x partial
      xbAcc[r]   = fmaf(xq, bc[col], xbAcc[r]);    // x . (A mu) partial
    }
  }
  const float common = p->common_, halfcp = p->halfcp;
  const float biasc = p->bias_const - 0.5f * logdet[c];
  const float ccc = ccv[c];
  for (int r = 0; r < 8; ++r) {
    float dxa = half16_reduce(distAcc[r]);
    float dxb = half16_reduce(xbAcc[r]);
    if (lo == 0) {
      int q = q0 + r + 8 * hi;
      float dist = dxa - 2.0f * dxb + ccc;
      out[(size_t)q * C + c] = biasc - halfcp * log1pf(dist / common);
    }
  }
}

// ---------------------------------------------------------------------------
extern "C" void kernel_launch(void* const* d_in, const int* in_sizes, int n_in,
                              void* d_out, int out_size, void* d_ws, size_t ws_size,
                              hipStream_t stream) {
  (void)n_in; (void)ws_size;
  const float* Xs    = (const float*)d_in[0];
  const int*   lab   = (const int*)d_in[1];
  const float* Xq    = (const float*)d_in[2];
  const float* m     = (const float*)d_in[3];
  const float* kappa = (const float*)d_in[4];
  const float* nu    = (const float*)d_in[5];
  const float* diag  = (const float*)d_in[6];
  const float* tl    = (const float*)d_in[7];

  const int D = in_sizes[6];          // triu_diag has D elements
  const int N = in_sizes[0] / D;      // support rows
  const int Q = in_sizes[2] / D;      // query rows
  const int C = out_size / Q;         // classes (logits are [Q, C])
  const int S = N / C;                // shots (balanced episode)
  float* out = (float*)d_out;

  // workspace carve-up (all fp32-aligned)
  float* wsf    = (float*)d_ws;
  Params* prm   = (Params*)wsf;                      // 16 floats
  float* mu     = wsf + 16;                          // C*D
  float* bvec   = mu + (size_t)C * D;                // C*D
  float* ccv    = bvec + (size_t)C * D;              // C
  float* logdet = ccv + C;                           // C
  int*   pos    = (int*)(logdet + C);                // N
  float* Xg     = (float*)(pos + N);                 // N*D
  float* base   = Xg + (size_t)N * D;                // D*D
  float* sigA   = base + (size_t)D * D;              // C*D*D (sigma -> R -> A)
  float* W      = sigA + (size_t)C * D * D;          // C*D*D

  k_params<<<1, 1, 0, stream>>>(kappa, nu, prm, D, S);
  k_pos<<<(N + 255) / 256, 256, 0, stream>>>(lab, pos, N);
  k_gather<<<N, 256, 0, stream>>>(Xs, lab, pos, Xg, D, S);
  k_mu<<<C, 256, 0, stream>>>(Xs, lab, m, prm, mu, N, D, S);
  k_base<<<(D * D + 255) / 256, 256, 0, stream>>>(diag, tl, m, prm, base, D);

  const int tilesPerClass = (D / 16) * (D / 16);
  const int gramWaves = C * tilesPerClass;
  const int gramBlocks = (gramWaves + 7) / 8;        // 8 waves / 256-thread block
  k_gram_sigma<<<gramBlocks, 256, 0, stream>>>(Xg, base, mu, prm, sigA, D, S, C);

  k_chol<<<C, 256, 0, stream>>>(sigA, logdet, D);
  k_trinv<<<C, 256, 0, stream>>>(sigA, W, D);
  k_wtw_reg<<<gramBlocks, 256, 0, stream>>>(W, sigA, D, C);   // sigA now holds A

  k_bvec<<<C, 256, 0, stream>>>(sigA, mu, bvec, D);
  k_cc<<<C, 256, 0, stream>>>(bvec, mu, ccv, D);

  // mahal: grid = (query strips, class groups of 8 waves)
  const int strip = 16 * D;
  const size_t lds_bytes = (size_t)(strip + 2 * (strip >> 8)) * sizeof(float);
  dim3 mgrid(Q / 16, (C + 7) / 8);
  k_mahal<<<mgrid, 256, lds_bytes, stream>>>(Xq, sigA, bvec, ccv, logdet, prm,
                                             out, D, C, Q);
}